// mk1_MuonEncoder_46566035423747
// MI455X (gfx1250) — compile-verified
//
#include <hip/hip_runtime.h>
#include <hip/hip_bf16.h>

typedef __attribute__((ext_vector_type(2))) float v2f;
typedef __attribute__((ext_vector_type(8))) float v8f;

#define NNODES 50000
#define NEDGES 800000

__device__ __forceinline__ float gelu_erf(float x) {
    return 0.5f * x * (1.0f + erff(x * 0.70710678118654752440f));
}

__device__ __forceinline__ void atomic_add_f32(float* p, float v) {
    unsafeAtomicAdd(p, v);   // native global_atomic_add_f32 on CDNA
}

// ---------------------------------------------------------------------------
// Generic zero
// ---------------------------------------------------------------------------
__global__ void zero_f32(float* p, long n) {
    long i = (long)blockIdx.x * blockDim.x + threadIdx.x;
    long stride = (long)gridDim.x * blockDim.x;
    for (; i < n; i += stride) p[i] = 0.0f;
}

// ---------------------------------------------------------------------------
// LayerNorm over 64 features, one wave per node (wave32)
// ---------------------------------------------------------------------------
__global__ void ln_kernel(const float* __restrict__ x, const float* __restrict__ g,
                          const float* __restrict__ b, float* __restrict__ out, int N) {
    int lane = threadIdx.x & 31;
    int wave = threadIdx.x >> 5;
    int n = blockIdx.x * (blockDim.x >> 5) + wave;
    if (n >= N) return;
    float v0 = x[n * 64 + lane];
    float v1 = x[n * 64 + 32 + lane];
    float s = v0 + v1, sq = v0 * v0 + v1 * v1;
    for (int m = 16; m >= 1; m >>= 1) {
        s  += __shfl_xor(s,  m, 32);
        sq += __shfl_xor(sq, m, 32);
    }
    float mu  = s * (1.0f / 64.0f);
    float var = sq * (1.0f / 64.0f) - mu * mu;   // biased var, matches jnp
    float rs  = rsqrtf(var + 1e-5f);
    out[n * 64 + lane]      = (v0 - mu) * rs * g[lane]      + b[lane];
    out[n * 64 + 32 + lane] = (v1 - mu) * rs * g[lane + 32] + b[lane + 32];
}

// ---------------------------------------------------------------------------
// Pack B[K,Nd] (row-major) into WMMA-fragment order, zero-padded to Ndp cols:
//   PB[((kt*Ndp + col)*2 + hi)*2 + {0,1}] = B[(kt*4 + hi*2 + {0,1})*Nd + col]
// so each lane's (b.x,b.y) is one aligned 8-byte load; per-kt stride = Ndp*4.
// ---------------------------------------------------------------------------
__global__ void pack_b(const float* __restrict__ B, float* __restrict__ PB,
                       int K, int Nd, int Ndp) {
    long total = (long)(K / 4) * Ndp * 2;
    long i = (long)blockIdx.x * blockDim.x + threadIdx.x;
    if (i >= total) return;
    int hi = (int)(i & 1);
    long rest = i >> 1;
    int col = (int)(rest % Ndp);
    int kt  = (int)(rest / Ndp);
    int k0 = kt * 4 + hi * 2;
    float v0 = 0.0f, v1 = 0.0f;
    if (col < Nd) {
        v0 = B[(size_t)k0 * Nd + col];
        v1 = B[(size_t)(k0 + 1) * Nd + col];
    }
    PB[i * 2 + 0] = v0;
    PB[i * 2 + 1] = v1;
}

// ---------------------------------------------------------------------------
// WMMA fp32 GEMM: Cout[M,Nd] = epilogue(A[M,K] @ B[K,Nd]); B prepacked (Ndp).
//   MODE 0: raw (+bias)   MODE 1: gelu(acc+bias)
//   MODE 2: gelu((acc+bias+Cin)*scale)   MODE 3: tanh(acc+bias)
// One wave per 16x16 tile; 8 waves/block -> 128 rows x 16 cols per block.
// A-frag (16x4 f32): lanes 0-15 -> K=(kk,kk+1), lanes 16-31 -> K=(kk+2,kk+3).
// ---------------------------------------------------------------------------
template <int MODE>
__global__ __launch_bounds__(256) void gemm_wmma(
    const float* __restrict__ A, const float* __restrict__ PB,
    const float* __restrict__ bias, const float* Cin, float* Cout,
    int M, int K, int Nd, int Ndp, float scale) {
    const int lane = threadIdx.x & 31;
    const int wave = threadIdx.x >> 5;
    const int row0 = (blockIdx.x * 8 + wave) * 16;
    const int col0 = blockIdx.y * 16;
    if (row0 >= M) return;                 // uniform per wave -> EXEC stays all-1s
    const int r  = lane & 15;
    const int hi = lane >> 4;              // 0 or 1
    const int arow = row0 + r;

    const float* ap = A + (size_t)arow * K + hi * 2;
    const float* bp = PB + ((size_t)(col0 + r) * 2 + hi) * 2;
    const size_t bstride = (size_t)Ndp * 4;

    v8f acc = {};
    const int KT = K >> 2;
    for (int kt = 0; kt < KT; ++kt) {
        v2f a = *reinterpret_cast<const v2f*>(ap);
        v2f b = *reinterpret_cast<const v2f*>(bp);
        ap += 4;
        bp += bstride;
        acc = __builtin_amdgcn_wmma_f32_16x16x4_f32(
            false, a, false, b, (short)0, acc, false, false);
    }

    const int ocol = col0 + r;
    const int ocolc = ocol < Nd ? ocol : Nd - 1;
    const float bv = bias ? bias[ocolc] : 0.0f;
    if (ocol < Nd) {
#pragma unroll
        for (int rr = 0; rr < 8; ++rr) {
            int orow = row0 + rr + hi * 8;
            float v = acc[rr] + bv;
            if (MODE == 2) v = (v + Cin[(size_t)orow * Nd + ocol]) * scale;
            if (MODE == 1 || MODE == 2) v = gelu_erf(v);
            if (MODE == 3) v = tanhf(v);
            Cout[(size_t)orow * Nd + ocol] = v;
        }
    }
}

// ---------------------------------------------------------------------------
// Degree counts / inverse
// ---------------------------------------------------------------------------
__global__ void count_edges(const int* __restrict__ EI, float* cnt, int E, int N) {
    long i = (long)blockIdx.x * blockDim.x + threadIdx.x;
    if (i >= 3L * E) return;
    int t = (int)(i / E), e = (int)(i % E);
    int dst = EI[(size_t)t * 2 * E + E + e];
    atomic_add_f32(&cnt[(size_t)t * N + dst], 1.0f);
}

__global__ void invert_cnt(float* cnt, long n) {
    long i = (long)blockIdx.x * blockDim.x + threadIdx.x;
    if (i < n) cnt[i] = 1.0f / fmaxf(cnt[i], 1.0f);
}

// ---------------------------------------------------------------------------
// SAGE gather + scatter-add: agg[dst, t*128 + c] += x[src, c]
// one thread per (t, e, 4-channel group)
// ---------------------------------------------------------------------------
__global__ void sage_scatter(const float* __restrict__ x, const int* __restrict__ EI,
                             float* agg, int E, long total) {
    long i = (long)blockIdx.x * blockDim.x + threadIdx.x;
    if (i >= total) return;
    int g = (int)(i & 31);
    long rest = i >> 5;
    int e = (int)(rest % E);
    int t = (int)(rest / E);
    int src = EI[(size_t)t * 2 * E + e];
    int dst = EI[(size_t)t * 2 * E + E + e];
    const float4 v = *reinterpret_cast<const float4*>(x + (size_t)src * 128 + g * 4);
    float* base = agg + (size_t)dst * 384 + t * 128 + g * 4;
    atomic_add_f32(base + 0, v.x);
    atomic_add_f32(base + 1, v.y);
    atomic_add_f32(base + 2, v.z);
    atomic_add_f32(base + 3, v.w);
}

__global__ void scale_agg(float* agg, const float* __restrict__ inv, int N) {
    long i = (long)blockIdx.x * blockDim.x + threadIdx.x;
    if (i >= (long)N * 384) return;
    int n = (int)(i / 384);
    int t = (int)((i % 384) >> 7);
    agg[i] *= inv[(size_t)t * N + n];
}

// ---------------------------------------------------------------------------
// GraphNorm: 3-pass column (128-wide) reduction
// ---------------------------------------------------------------------------
__global__ void gn_pass1(const float* __restrict__ y, float* stats, int rows) {
    int c = threadIdx.x;
    float s = 0.0f;
    for (int r = blockIdx.x; r < rows; r += gridDim.x) s += y[(size_t)r * 128 + c];
    atomic_add_f32(&stats[c], s);
}

__global__ void gn_pass2(float* y, float* stats, const float* __restrict__ scale,
                         int rows, float invN) {
    int c = threadIdx.x;
    float m = stats[c] * invN * scale[c];
    float s = 0.0f;
    for (int r = blockIdx.x; r < rows; r += gridDim.x) {
        float t = y[(size_t)r * 128 + c] - m;
        y[(size_t)r * 128 + c] = t;
        s += t * t;
    }
    atomic_add_f32(&stats[128 + c], s);
}

__global__ void gn_pass3(const float* __restrict__ y, const float* __restrict__ stats,
                         const float* __restrict__ w, const float* __restrict__ b,
                         float* __restrict__ x, float* __restrict__ xsave,
                         int rows, float invN, int layer) {
    long i = (long)blockIdx.x * blockDim.x + threadIdx.x;
    if (i >= (long)rows * 128) return;
    int c = (int)(i & 127);
    long n = i >> 7;
    float var = stats[128 + c] * invN;
    float v = y[i] * rsqrtf(var + 1e-5f) * w[c] + b[c];
    x[i] = v;
    xsave[n * 384 + layer * 128 + c] = v;
}

// ---------------------------------------------------------------------------
// Weight prep: sum Wr over edge types, sum bias over edge types
// ---------------------------------------------------------------------------
__global__ void prep_wrsum(const float* __restrict__ Wr, float* out) {
    int i = blockIdx.x * blockDim.x + threadIdx.x;
    if (i >= 3 * 16384) return;
    int l = i >> 14, rc = i & 16383;
    out[i] = Wr[(size_t)(l * 3 + 0) * 16384 + rc] +
             Wr[(size_t)(l * 3 + 1) * 16384 + rc] +
             Wr[(size_t)(l * 3 + 2) * 16384 + rc];
}

__global__ void prep_bsum(const float* __restrict__ bs, float* out) {
    int i = blockIdx.x * blockDim.x + threadIdx.x;
    if (i >= 3 * 128) return;
    int l = i >> 7, c = i & 127;
    out[i] = bs[(l * 3 + 0) * 128 + c] + bs[(l * 3 + 1) * 128 + c] + bs[(l * 3 + 2) * 128 + c];
}

// ---------------------------------------------------------------------------
// Concat [e2 | x_AA] -> [N, 120]
// ---------------------------------------------------------------------------
__global__ void concat_aa(const float* __restrict__ e2, const float* __restrict__ xaa,
                          float* __restrict__ xcat, int N) {
    long i = (long)blockIdx.x * blockDim.x + threadIdx.x;
    if (i >= (long)N * 120) return;
    int c = (int)(i % 120);
    long n = i / 120;
    xcat[i] = (c < 100) ? e2[n * 100 + c] : xaa[n * 20 + (c - 100)];
}

// ---------------------------------------------------------------------------
// Vector quantizer: codebook in LDS, per-thread argmin (first-min tie-break)
// ---------------------------------------------------------------------------
__global__ __launch_bounds__(256) void vq_kernel(
    const float* __restrict__ xf, const float* __restrict__ cb,
    float* __restrict__ zq, float* loss_acc, int N) {
    __shared__ float scb[128 * 32];
    for (int i = threadIdx.x; i < 128 * 32; i += blockDim.x) scb[i] = cb[i];
    __syncthreads();
    int n = blockIdx.x * blockDim.x + threadIdx.x;
    float lsum = 0.0f;
    if (n < N) {
        float v[32];
#pragma unroll
        for (int j = 0; j < 32; ++j) v[j] = xf[(size_t)n * 32 + j];
        float best = 3.4e38f;
        int bi = 0;
        for (int k = 0; k < 128; ++k) {
            float d = 0.0f;
#pragma unroll
            for (int j = 0; j < 32; ++j) {
                float t = v[j] - scb[k * 32 + j];
                d += t * t;
            }
            if (d < best) { best = d; bi = k; }
        }
#pragma unroll
        for (int j = 0; j < 32; ++j) {
            float q = scb[bi * 32 + j];
            zq[(size_t)n * 32 + j] = q;   // z_q = x + sg(q - x) == q in forward
            float t = q - v[j];
            lsum += t * t;
        }
    }
    __shared__ float red[256];
    red[threadIdx.x] = lsum;
    __syncthreads();
    for (int s = 128; s > 0; s >>= 1) {
        if ((int)threadIdx.x < s) red[threadIdx.x] += red[threadIdx.x + s];
        __syncthreads();
    }
    if (threadIdx.x == 0) atomic_add_f32(loss_acc, red[0]);
}

__global__ void vq_finalize(const float* loss_acc, float* out, float scale) {
    if (threadIdx.x == 0 && blockIdx.x == 0) out[0] = loss_acc[0] * scale;
}

// ---------------------------------------------------------------------------
// Host side
// ---------------------------------------------------------------------------
extern "C" void kernel_launch(void* const* d_in, const int* in_sizes, int n_in,
                              void* d_out, int out_size, void* d_ws, size_t ws_size,
                              hipStream_t stream) {
    (void)in_sizes; (void)n_in; (void)out_size; (void)ws_size;
    const float* x_res = (const float*)d_in[0];
    const float* x_AA  = (const float*)d_in[1];
    const int*   EI    = (const int*)d_in[2];
    const float* ln_g  = (const float*)d_in[3];
    const float* ln_b  = (const float*)d_in[4];
    const float* W1    = (const float*)d_in[5];
    const float* b1    = (const float*)d_in[6];
    const float* W2    = (const float*)d_in[7];
    const float* b2    = (const float*)d_in[8];
    const float* sWl   = (const float*)d_in[9];
    const float* sWr   = (const float*)d_in[10];
    const float* sb    = (const float*)d_in[11];
    const float* gnw   = (const float*)d_in[12];
    const float* gnb   = (const float*)d_in[13];
    const float* gns   = (const float*)d_in[14];
    const float* Wh1   = (const float*)d_in[15];
    const float* bh1   = (const float*)d_in[16];
    const float* Wh2   = (const float*)d_in[17];
    const float* bh2   = (const float*)d_in[18];
    const float* Wo1   = (const float*)d_in[19];
    const float* bo1   = (const float*)d_in[20];
    const float* Wo2   = (const float*)d_in[21];
    const float* bo2   = (const float*)d_in[22];
    const float* Wo3   = (const float*)d_in[23];
    const float* bo3   = (const float*)d_in[24];
    const float* cb    = (const float*)d_in[25];

    const int N = NNODES, E = NEDGES;
    float* ws = (float*)d_ws;
    // workspace plan (floats); big buffers are reused across phases
    size_t o_agg  = 0;                               // N*384: h1 / agg / e1,e2,xcat
    size_t o_save = o_agg  + (size_t)N * 384;        // N*384: JK concat
    size_t o_x    = o_save + (size_t)N * 384;        // N*128: x / o2
    size_t o_tmp  = o_x    + (size_t)N * 128;        // N*128: xr / sage tmp / o1
    size_t o_fin  = o_tmp  + (size_t)N * 128;        // N*32 : head output
    size_t o_inv  = o_fin  + (size_t)N * 32;         // 3*N  : 1/deg
    size_t o_wrs  = o_inv  + (size_t)3 * N;          // 3*128*128
    size_t o_bs   = o_wrs  + (size_t)3 * 128 * 128;  // 3*128
    size_t o_st   = o_bs   + (size_t)3 * 128;        // 256 (colsum | colvar)
    size_t o_loss = o_st   + 256;                    // 1
    // packed weights
    size_t o_pW1  = o_loss + 1;                      // 64*256
    size_t o_pW2  = o_pW1  + (size_t)64 * 256;       // 256*128
    size_t o_pWl  = o_pW2  + (size_t)256 * 128;      // 1152*128  (3 layers, each 384*128)
    size_t o_pWr  = o_pWl  + (size_t)1152 * 128;     // 384*128   (3 layers, each 128*128)
    size_t o_pWh1 = o_pWr  + (size_t)384 * 128;      // 384*112
    size_t o_pWh2 = o_pWh1 + (size_t)384 * 112;      // 100*112
    size_t o_pWo1 = o_pWh2 + (size_t)100 * 112;      // 120*112
    size_t o_pWo2 = o_pWo1 + (size_t)120 * 112;      // 100*112
    size_t o_pWo3 = o_pWo2 + (size_t)100 * 112;      // 100*32

    float* agg   = ws + o_agg;
    float* xsave = ws + o_save;
    float* x     = ws + o_x;
    float* tmp   = ws + o_tmp;
    float* xfin  = ws + o_fin;
    float* inv   = ws + o_inv;
    float* wrs   = ws + o_wrs;
    float* bsum  = ws + o_bs;
    float* stats = ws + o_st;
    float* loss  = ws + o_loss;
    float* pW1   = ws + o_pW1;
    float* pW2   = ws + o_pW2;
    float* pWl   = ws + o_pWl;
    float* pWr   = ws + o_pWr;
    float* pWh1  = ws + o_pWh1;
    float* pWh2  = ws + o_pWh2;
    float* pWo1  = ws + o_pWo1;
    float* pWo2  = ws + o_pWo2;
    float* pWo3  = ws + o_pWo3;
    float* zq    = (float*)d_out;

    const int GX = (N + 127) / 128;    // 391 row-blocks for WMMA GEMMs
    auto pk = [](int K, int Ndp) { return (int)(((long)(K / 4) * Ndp * 2 + 255) / 256); };

    // ---- prep: folded weights + packed WMMA B-fragments ----
    prep_wrsum<<<(3 * 16384 + 255) / 256, 256, 0, stream>>>(sWr, wrs);
    prep_bsum<<<2, 256, 0, stream>>>(sb, bsum);
    pack_b<<<pk(64, 256),  256, 0, stream>>>(W1,  pW1,  64,   256, 256);
    pack_b<<<pk(256, 128), 256, 0, stream>>>(W2,  pW2,  256,  128, 128);
    pack_b<<<pk(1152,128), 256, 0, stream>>>(sWl, pWl,  1152, 128, 128);
    pack_b<<<pk(384, 128), 256, 0, stream>>>(wrs, pWr,  384,  128, 128);
    pack_b<<<pk(384, 112), 256, 0, stream>>>(Wh1, pWh1, 384,  100, 112);
    pack_b<<<pk(100, 112), 256, 0, stream>>>(Wh2, pWh2, 100,  100, 112);
    pack_b<<<pk(120, 112), 256, 0, stream>>>(Wo1, pWo1, 120,  100, 112);
    pack_b<<<pk(100, 112), 256, 0, stream>>>(Wo2, pWo2, 100,  100, 112);
    pack_b<<<pk(100, 32),  256, 0, stream>>>(Wo3, pWo3, 100,  32,  32);
    zero_f32<<<1024, 256, 0, stream>>>(inv, 3L * N);
    count_edges<<<(3 * E + 255) / 256, 256, 0, stream>>>(EI, inv, E, N);
    invert_cnt<<<(3 * N + 255) / 256, 256, 0, stream>>>(inv, 3L * N);

    // ---- input block: LayerNorm + MLP 64 -> 256 -> 128 ----
    ln_kernel<<<(N + 7) / 8, 256, 0, stream>>>(x_res, ln_g, ln_b, tmp, N);
    gemm_wmma<1><<<dim3(GX, 16), 256, 0, stream>>>(tmp, pW1, b1, nullptr, agg, N, 64, 256, 256, 1.0f);
    gemm_wmma<1><<<dim3(GX, 8), 256, 0, stream>>>(agg, pW2, b2, nullptr, x, N, 256, 128, 128, 1.0f);

    // ---- 3 SAGE layers ----
    const long scatter_total = 3L * E * 32;
    for (int l = 0; l < 3; ++l) {
        zero_f32<<<4096, 256, 0, stream>>>(agg, (long)N * 384);
        sage_scatter<<<(int)((scatter_total + 255) / 256), 256, 0, stream>>>(x, EI, agg, E, scatter_total);
        scale_agg<<<(int)(((long)N * 384 + 255) / 256), 256, 0, stream>>>(agg, inv, N);
        // tmp = [agg0|agg1|agg2] @ vstack(Wl[l])      (K = 384)
        gemm_wmma<0><<<dim3(GX, 8), 256, 0, stream>>>(
            agg, pWl + (size_t)l * 384 / 4 * 128 * 4, nullptr, nullptr, tmp, N, 384, 128, 128, 1.0f);
        // tmp = gelu((x @ sum_t Wr + sum_t b + tmp) / 3)
        gemm_wmma<2><<<dim3(GX, 8), 256, 0, stream>>>(
            x, pWr + (size_t)l * 128 / 4 * 128 * 4, bsum + l * 128, tmp, tmp, N, 128, 128, 128, 1.0f / 3.0f);
        // GraphNorm -> x (next layer input) and xsave[:, l*128:(l+1)*128]
        zero_f32<<<1, 256, 0, stream>>>(stats, 256);
        gn_pass1<<<512, 128, 0, stream>>>(tmp, stats, N);
        gn_pass2<<<512, 128, 0, stream>>>(tmp, stats, gns + l * 128, N, 1.0f / (float)N);
        gn_pass3<<<(int)(((long)N * 128 + 255) / 256), 256, 0, stream>>>(
            tmp, stats, gnw + l * 128, gnb + l * 128, x, xsave, N, 1.0f / (float)N, l);
    }

    // ---- head ----
    float* e1   = agg;
    float* e2   = agg + (size_t)N * 100;
    float* xcat = agg + (size_t)N * 200;
    gemm_wmma<1><<<dim3(GX, 7), 256, 0, stream>>>(xsave, pWh1, bh1, nullptr, e1, N, 384, 100, 112, 1.0f);
    gemm_wmma<1><<<dim3(GX, 7), 256, 0, stream>>>(e1, pWh2, bh2, nullptr, e2, N, 100, 100, 112, 1.0f);
    concat_aa<<<(int)(((long)N * 120 + 255) / 256), 256, 0, stream>>>(e2, x_AA, xcat, N);
    gemm_wmma<1><<<dim3(GX, 7), 256, 0, stream>>>(xcat, pWo1, bo1, nullptr, tmp, N, 120, 100, 112, 1.0f);
    gemm_wmma<1><<<dim3(GX, 7), 256, 0, stream>>>(tmp, pWo2, bo2, nullptr, x, N, 100, 100, 112, 1.0f);
    gemm_wmma<3><<<dim3(GX, 2), 256, 0, stream>>>(x, pWo3, bo3, nullptr, xfin, N, 100, 32, 32, 1.0f);

    // ---- vector quantizer ----
    zero_f32<<<1, 64, 0, stream>>>(loss, 1);
    vq_kernel<<<(N + 255) / 256, 256, 0, stream>>>(xfin, cb, zq, loss, N);
    vq_finalize<<<1, 32, 0, stream>>>(loss, zq + (size_t)N * 32, 1.25f / ((float)N * 32.0f));
}